// DynamicMaskHead_28767690948890
// MI455X (gfx1250) — compile-verified
//
#include <hip/hip_runtime.h>
#include <stdint.h>

// ---------------- problem constants (from setup_inputs) ----------------
#define CIN   8          // mask feat channels
#define CH    8          // hidden channels
#define K0IN  10         // CIN + 2 (rel coords)
#define HH    112
#define WW    200
#define HWSZ  (HH*WW)
#define NPAR  169        // per-instance param vector length
#define FACTOR 2         // stride(8) / MASK_OUT_STRIDE(4)
#define OHH   (HH*FACTOR)   // 224
#define OWW   (WW*FACTOR)   // 400

// tile of logits computed per block: (TH+1) x (TW+1) positions incl. top/left halo
#define TH 7
#define TW 31
// (TH+1)*(TW+1) == 256 positions == 8 waves * 2 groups * 16 positions

// LDS feature tile: [8 ch][8 rows][32 cols], row pitch 33 dwords (TDM pad)
#define FPITCH 33
#define FCHP   (8*FPITCH)          // 264 dwords per channel
#define FTOT   (CIN*FCHP + 40)     // cushion for the +1 row/col halo shift

typedef float v2f __attribute__((ext_vector_type(2)));
typedef float v8f __attribute__((ext_vector_type(8)));
typedef unsigned int v4u __attribute__((ext_vector_type(4)));
typedef int v4i __attribute__((ext_vector_type(4)));
typedef int v8i __attribute__((ext_vector_type(8)));

#if __has_builtin(__builtin_amdgcn_wmma_f32_16x16x4_f32)
#define HAVE_WMMA_F32X4 1
#endif
#if __has_builtin(__builtin_amdgcn_tensor_load_to_lds)
#define HAVE_TDM 1
#endif

// D(16x16,f32) = A(16x4,f32) * B(4x16,f32) + C   (layouts per ISA 7.12.2)
__device__ __forceinline__ v8f wmma4(v2f a, v2f b, v8f c) {
#ifdef HAVE_WMMA_F32X4
  return __builtin_amdgcn_wmma_f32_16x16x4_f32(false, a, false, b, (short)0, c,
                                               false, false);
#else
  const int lane = __lane_id();
  const int n = lane & 15;
  #pragma unroll
  for (int r = 0; r < 8; ++r) {
    const int M = (lane < 16) ? r : (r + 8);
    float sum = c[r];
    #pragma unroll
    for (int k = 0; k < 4; ++k) {
      const int half = (k >> 1) * 16;
      float av = __shfl((k & 1) ? a.y : a.x, (M & 15) + half, 32);
      float bv = __shfl((k & 1) ? b.y : b.x, n + half, 32);
      sum += av * bv;
    }
    c[r] = sum;
  }
  return c;
#endif
}

__device__ __forceinline__ float chanlds(int ch, float relx, float rely,
                                         const float* sF, int tyx) {
  if (ch == 0) return relx;
  if (ch == 1) return rely;
  if (ch < K0IN) return sF[(ch - 2) * FCHP + tyx];
  return 0.0f;
}

__global__ __launch_bounds__(256) void dmh_fused_kernel(
    const float* __restrict__ feats,     // [N,8,H,W]
    const float* __restrict__ params,    // [n_inst,169]
    const float* __restrict__ locs,      // [n_inst,2]
    const float* __restrict__ soi,       // [5]
    const int*  __restrict__ im_inds,    // [n_inst]
    const int*  __restrict__ fpn,        // [n_inst]
    const int*  __restrict__ stride_p,   // [1]
    float* __restrict__ out)             // [n_inst,224,400]
{
  __shared__ float sP[NPAR + 7];
  __shared__ float sMeta[4];
  __shared__ float sStage[8][CH][16];       // per-wave D->B relayout buffer
  __shared__ float sLog[TH + 1][TW + 2];    // logit tile incl. halo (padded)
  __shared__ float sFeat[FTOT];             // feature tile staged by TDM

  const int inst = blockIdx.z;
  const int h0 = blockIdx.y * TH;
  const int w0 = blockIdx.x * TW;
  const int tid = threadIdx.x;
  const int lane = tid & 31;
  const int wv = tid >> 5;

  const int base_r = h0 - 1, base_c = w0 - 1;
  const float* fbase = feats + (size_t)im_inds[inst] * (CIN * HWSZ);

#ifdef HAVE_TDM
  // ---- one TDM descriptor loads the whole 8ch x 8row x 32col f32 tile ----
  if (wv == 0) {
    const int r0 = base_r < 0 ? 0 : base_r;      // shift at top/left image edge
    const int c0 = base_c < 0 ? 0 : base_c;      // (shifted halo slots unused)
    const uint32_t rsh = (base_r < 0) ? 1u : 0u;
    const uint32_t csh = (base_c < 0) ? 1u : 0u;
    const uint64_t ga =
        (uint64_t)(uintptr_t)fbase + 4ull * ((uint64_t)r0 * WW + (uint64_t)c0);
    const uint32_t lds0 =
        (uint32_t)(uintptr_t)(&sFeat[0]) + 4u * (rsh * FPITCH + csh);
    const uint32_t td0 = (uint32_t)(WW - c0);    // x extent from tile start
    const uint32_t td1 = (uint32_t)(HH - r0);    // y extent from tile start
    const uint64_t s0 = (uint64_t)WW;            // row stride (elements)
    const uint64_t s1 = (uint64_t)HWSZ;          // channel-plane stride

    v4u g0;
    g0[0] = 1u;                                   // count=1, user descriptor
    g0[1] = lds0;                                 // lds_addr (bytes)
    g0[2] = (uint32_t)ga;                         // global_addr[31:0]
    g0[3] = (uint32_t)((ga >> 32) & 0x1FFFFFFu) | (2u << 30); // [56:32] | type=2

    v8i g1;
    g1[0] = (int)((2u << 16)      // data_size = 4B
                | (1u << 20)      // pad_enable
                | (4u << 22));    // pad_interval = 32 dwords
                                  // pad_amount  = 1 dword (code 0)
    g1[1] = (int)((td0 & 0xFFFFu) << 16);
    g1[2] = (int)((td0 >> 16) | ((td1 & 0xFFFFu) << 16));
    g1[3] = (int)((td1 >> 16) | (32u << 16));     // tile_dim0 = 32
    g1[4] = (int)(8u | (8u << 16));               // tile_dim1 = 8, tile_dim2 = 8
    g1[5] = (int)(uint32_t)(s0 & 0xFFFFFFFFu);
    g1[6] = (int)((uint32_t)(s0 >> 32) | (((uint32_t)s1 & 0xFFFFu) << 16));
    g1[7] = (int)(uint32_t)(s1 >> 16);

    v4i g2; g2[0] = CIN; g2[1] = 0; g2[2] = 0; g2[3] = 0;  // tensor_dim2 = 8
    v4i g3; g3[0] = 0;   g3[1] = 0; g3[2] = 0; g3[3] = 0;

#if defined(__clang_major__) && (__clang_major__ >= 23)
    v8i gz; gz[0]=0; gz[1]=0; gz[2]=0; gz[3]=0; gz[4]=0; gz[5]=0; gz[6]=0; gz[7]=0;
    __builtin_amdgcn_tensor_load_to_lds(g0, g1, g2, g3, gz, 0);
#else
    __builtin_amdgcn_tensor_load_to_lds(g0, g1, g2, g3, 0);
#endif
  }
#endif

  if (tid < NPAR) sP[tid] = params[inst * NPAR + tid];
  if (tid == 0) {
    sMeta[0] = locs[inst * 2 + 0];
    sMeta[1] = locs[inst * 2 + 1];
    sMeta[2] = 1.0f / soi[fpn[inst]];
    sMeta[3] = (float)stride_p[0];
  }

#ifdef HAVE_TDM
  if (wv == 0) {
#if __has_builtin(__builtin_amdgcn_s_wait_tensorcnt)
    __builtin_amdgcn_s_wait_tensorcnt(0);
#else
    asm volatile("s_wait_tensorcnt 0x0" ::: "memory");
#endif
  }
#else
  // cooperative fallback: fill the same LDS layout with clamped loads
  for (int idx = tid; idx < CIN * 8 * 32; idx += 256) {
    const int z = idx >> 8;
    const int ry = (idx >> 5) & 7;
    const int cx = idx & 31;
    int gr = base_r + ry; gr = gr < 0 ? 0 : (gr > HH - 1 ? HH - 1 : gr);
    int gc = base_c + cx; gc = gc < 0 ? 0 : (gc > WW - 1 ? WW - 1 : gc);
    sFeat[z * FCHP + ry * FPITCH + cx] = fbase[z * HWSZ + gr * WW + gc];
  }
#endif
  __syncthreads();

  const float instx = sMeta[0], insty = sMeta[1], inv_soi = sMeta[2];
  const float strf = sMeta[3], half_str = 0.5f * strf;

  // ---- A fragments (weights), built once per block ----
  const int m = lane & 15;
  const int khalf = (lane < 16) ? 0 : 2;
  v2f a0[3], a1[2];
  #pragma unroll
  for (int c = 0; c < 3; ++c) {        // layer0 w0: [8][10], K padded to 12
    const int kb = 4 * c + khalf;
    v2f a; a.x = 0.0f; a.y = 0.0f;
    if (m < CH) {
      if (kb     < K0IN) a.x = sP[m * K0IN + kb];
      if (kb + 1 < K0IN) a.y = sP[m * K0IN + kb + 1];
    }
    a0[c] = a;
  }
  #pragma unroll
  for (int c = 0; c < 2; ++c) {        // layer1 w1: [8][8]
    const int kb = 4 * c + khalf;
    v2f a; a.x = 0.0f; a.y = 0.0f;
    if (m < CH) { a.x = sP[80 + m * CH + kb]; a.y = sP[80 + m * CH + kb + 1]; }
    a1[c] = a;
  }

  // ---- compute logits for the (TH+1)x(TW+1) tile; 16 positions per group ----
  #pragma unroll
  for (int gi = 0; gi < 2; ++gi) {
    const int g = wv * 2 + gi;
    const int pos16 = lane & 15;
    const int p = g * 16 + pos16;            // 0..255 over 8x32 local tile
    const int ty = p >> 5, tx = p & 31;
    int pr = base_r + ty; pr = pr < 0 ? 0 : (pr > HH - 1 ? HH - 1 : pr);
    int pc = base_c + tx; pc = pc < 0 ? 0 : (pc > WW - 1 ? WW - 1 : pc);
    const int tyx = ty * FPITCH + tx;
    const float relx = (instx - ((float)pc * strf + half_str)) * inv_soi;
    const float rely = (insty - ((float)pr * strf + half_str)) * inv_soi;

    // layer 0: B = x[10 x 16], 3 K-chunks, features from LDS tile
    v2f b0f[3];
    #pragma unroll
    for (int c = 0; c < 3; ++c) {
      const int kb = 4 * c + khalf;
      v2f b;
      b.x = chanlds(kb,     relx, rely, sFeat, tyx);
      b.y = chanlds(kb + 1, relx, rely, sFeat, tyx);
      b0f[c] = b;
    }
    v8f acc = {0.f, 0.f, 0.f, 0.f, 0.f, 0.f, 0.f, 0.f};
    acc = wmma4(a0[0], b0f[0], acc);
    acc = wmma4(a0[1], b0f[1], acc);
    acc = wmma4(a0[2], b0f[2], acc);

    // bias + relu, relayout D (C-layout) -> B-layout via per-wave LDS staging
    if (lane < 16) {
      #pragma unroll
      for (int ch = 0; ch < CH; ++ch) {
        float v = acc[ch] + sP[152 + ch];
        sStage[wv][ch][pos16] = v > 0.0f ? v : 0.0f;
      }
    }
    __builtin_amdgcn_wave_barrier();   // LDS is in-order within a wave
    v2f b1f[2];
    #pragma unroll
    for (int c = 0; c < 2; ++c) {
      const int kb = 4 * c + khalf;
      v2f b;
      b.x = sStage[wv][kb][pos16];
      b.y = sStage[wv][kb + 1][pos16];
      b1f[c] = b;
    }
    __builtin_amdgcn_wave_barrier();

    // layer 1: 2 K-chunks
    v8f acc1 = {0.f, 0.f, 0.f, 0.f, 0.f, 0.f, 0.f, 0.f};
    acc1 = wmma4(a1[0], b1f[0], acc1);
    acc1 = wmma4(a1[1], b1f[1], acc1);

    // layer 2: 8-wide dot in VALU, write logit tile
    if (lane < 16) {
      float logit = sP[168];
      #pragma unroll
      for (int ch = 0; ch < CH; ++ch) {
        float v = acc1[ch] + sP[160 + ch];
        v = v > 0.0f ? v : 0.0f;
        logit += sP[144 + ch] * v;
      }
      sLog[ty][tx] = logit;
    }
  }
  __syncthreads();

  // ---- fused aligned_bilinear x2: out[Y][X] = up[max(Y-1,0)][max(X-1,0)] ----
  float* obase = out + (size_t)inst * (OHH * OWW);
  const int OWt = 2 * TW;                       // 62
  for (int idx = tid; idx < 2 * TH * OWt; idx += 256) {
    const int y = idx / OWt, x = idx - y * OWt;
    const int Y = 2 * h0 + y, X = 2 * w0 + x;
    if (X < OWW && Y < OHH) {
      int i = Y - 1; i = i < 0 ? 0 : i;
      int j = X - 1; j = j < 0 ? 0 : j;
      const int iy = i >> 1, fy = i & 1;
      const int ix = j >> 1, fx = j & 1;
      const int ly = iy - base_r, lx = ix - base_c;   // in [0,TH]/[0,TW]
      const float v00 = sLog[ly][lx];
      const float v01 = sLog[ly][lx + fx];
      const float v10 = sLog[ly + fy][lx];
      const float v11 = sLog[ly + fy][lx + fx];
      const float tx_ = 0.5f * (float)fx, ty_ = 0.5f * (float)fy;
      const float top = v00 + tx_ * (v01 - v00);
      const float bot = v10 + tx_ * (v11 - v10);
      obase[(size_t)Y * OWW + X] = top + ty_ * (bot - top);
    }
  }
}

extern "C" void kernel_launch(void* const* d_in, const int* in_sizes, int n_in,
                              void* d_out, int out_size, void* d_ws, size_t ws_size,
                              hipStream_t stream) {
  (void)n_in; (void)out_size; (void)d_ws; (void)ws_size;
  const float* feats   = (const float*)d_in[0];
  const float* params  = (const float*)d_in[1];
  const float* locs    = (const float*)d_in[2];
  const float* soi     = (const float*)d_in[3];
  const int*   im_inds = (const int*)d_in[4];
  const int*   fpn     = (const int*)d_in[5];
  const int*   stridep = (const int*)d_in[6];
  const int n_inst = in_sizes[1] / NPAR;

  dim3 grid((WW + TW - 1) / TW, (HH + TH - 1) / TH, n_inst);
  dmh_fused_kernel<<<grid, 256, 0, stream>>>(feats, params, locs, soi,
                                             im_inds, fpn, stridep,
                                             (float*)d_out);
}